// GPSLayer_36558761624223
// MI455X (gfx1250) — compile-verified
//
#include <hip/hip_runtime.h>
#include <hip/hip_bf16.h>

typedef __attribute__((ext_vector_type(2))) float v2f;
typedef __attribute__((ext_vector_type(8))) float v8f;

#define D 64            // feature dim (in == out == 64)
#define BLK 256         // threads per block (8 wave32)
#define ROWS_PER_BLK 32 // 2 row tiles x 4 col tiles = 8 wave tiles
#define LDA 68          // LDS stride for A tile (float units)  -> conflict-free column reads
#define LDW 80          // LDS stride for W tile (float units)  -> disjoint banks for lane halves

// ---------------- degree / normalization ----------------

__global__ __launch_bounds__(256) void k_init_deg(float* deg, int n) {
    int i = blockIdx.x * blockDim.x + threadIdx.x;
    if (i < n) deg[i] = 1.0f;  // self loop
}

__global__ __launch_bounds__(256) void k_count_deg(const int* __restrict__ dst, float* deg, int e) {
    int i = blockIdx.x * blockDim.x + threadIdx.x;
    if (i < e) atomicAdd(&deg[dst[i]], 1.0f);
}

__global__ __launch_bounds__(256) void k_dinv(float* deg, int n) {
    int i = blockIdx.x * blockDim.x + threadIdx.x;
    if (i < n) deg[i] = 1.0f / sqrtf(deg[i]);  // deg >= 1 always (self loops)
}

// ---------------- WMMA GEMM (f32 16x16x4), two fused epilogues ----------------
// MODE 0:  hs = (A @ W) * dinv[row];  out0 = hs, out1 = acc (self-loop init)
// MODE 1:  out0[row] = accb[row]*dinv[row] + (A @ W) + bg[col] + bp[col]
template <int MODE>
__global__ __launch_bounds__(BLK) void k_gemm_wmma(
    const float* __restrict__ A,    // [N, 64]
    const float* __restrict__ W,    // [64, 64]
    const float* __restrict__ dinv, // [N]
    float* __restrict__ out0,
    float* __restrict__ out1,
    const float* __restrict__ accb,
    const float* __restrict__ bg,
    const float* __restrict__ bp)
{
    __shared__ float ldsA[ROWS_PER_BLK * LDA];
    __shared__ float ldsW[D * LDW];

    const int tid = threadIdx.x;
    const int gr0 = blockIdx.x * ROWS_PER_BLK;

    // Stage 32x64 A tile (coalesced float4): 512 float4 loads across 256 threads.
    {
        const float* ap = A + (size_t)gr0 * D;
        #pragma unroll
        for (int f = tid; f < (ROWS_PER_BLK * D) / 4; f += BLK) {
            int r  = f >> 4;           // 16 float4 per row
            int c4 = (f & 15) << 2;
            float4 v = *(const float4*)(ap + r * D + c4);
            *(float4*)(&ldsA[r * LDA + c4]) = v;
        }
    }
    // Stage 64x64 W tile: 1024 float4 loads.
    {
        #pragma unroll
        for (int f = tid; f < (D * D) / 4; f += BLK) {
            int r  = f >> 4;
            int c4 = (f & 15) << 2;
            float4 v = *(const float4*)(W + r * D + c4);
            *(float4*)(&ldsW[r * LDW + c4]) = v;
        }
    }
    __syncthreads();

    const int lane  = tid & 31;
    const int wv    = tid >> 5;
    const int half  = lane >> 4;       // 0: lanes 0-15, 1: lanes 16-31
    const int m     = lane & 15;
    const int rbase = (wv >> 2) << 4;  // 0 or 16
    const int cbase = (wv & 3) << 4;   // 0,16,32,48

    v8f acc = {};
    #pragma unroll
    for (int k0 = 0; k0 < D; k0 += 4) {
        const int ka = k0 + 2 * half;  // A/B vgpr j, lane-half h -> K = k0 + j + 2h
        v2f a = *(const v2f*)(&ldsA[(rbase + m) * LDA + ka]);  // 8B-aligned ds_load_b64
        v2f b;
        b.x = ldsW[(ka + 0) * LDW + cbase + m];
        b.y = ldsW[(ka + 1) * LDW + cbase + m];
        acc = __builtin_amdgcn_wmma_f32_16x16x4_f32(
            /*neg_a=*/false, a, /*neg_b=*/false, b,
            /*c_mod=*/(short)0, acc, /*reuse_a=*/false, /*reuse_b=*/false);
    }

    const int col = cbase + m;
    float bias = 0.0f;
    if (MODE == 1) bias = bg[col] + bp[col];

    // C/D layout: vgpr v -> row M = v + 8*half, col N = lane%16
    #pragma unroll
    for (int v = 0; v < 8; ++v) {
        const int row = gr0 + rbase + v + 8 * half;
        const size_t o = (size_t)row * D + col;
        const float di = dinv[row];
        if (MODE == 0) {
            float s = acc[v] * di;
            out0[o] = s;   // hs (gather source)
            out1[o] = s;   // acc init (self-loop contribution)
        } else {
            out0[o] = accb[o] * di + acc[v] + bias;
        }
    }
}

// Naive remainder path (only launched if N % 32 != 0; N=100000 -> unused)
template <int MODE>
__global__ __launch_bounds__(256) void k_gemm_naive(
    const float* __restrict__ A, const float* __restrict__ W,
    const float* __restrict__ dinv, float* out0, float* out1,
    const float* __restrict__ accb, const float* __restrict__ bg,
    const float* __restrict__ bp, int rowStart, int nRows)
{
    int idx = blockIdx.x * blockDim.x + threadIdx.x;
    if (idx >= nRows * D) return;
    int r = rowStart + idx / D;
    int c = idx % D;
    float s = 0.0f;
    for (int k = 0; k < D; ++k) s += A[(size_t)r * D + k] * W[k * D + c];
    size_t o = (size_t)r * D + c;
    float di = dinv[r];
    if (MODE == 0) { float v = s * di; out0[o] = v; out1[o] = v; }
    else           { out0[o] = accb[o] * di + s + bg[c] + bp[c]; }
}

// ---------------- edge scatter: acc[dst] += hs[src] ----------------
// One thread per (edge, column): coalesced reads of hs[src] row, f32 atomics.
__global__ __launch_bounds__(256) void k_scatter(
    const int* __restrict__ src, const int* __restrict__ dst,
    const float* __restrict__ hs, float* __restrict__ acc, int total /* = E*64 */)
{
    int idx = blockIdx.x * blockDim.x + threadIdx.x;
    if (idx >= total) return;
    int e = idx >> 6;
    int c = idx & 63;
    int s = src[e];
    int d = dst[e];
    atomicAdd(&acc[(size_t)d * D + c], hs[(size_t)s * D + c]);
}

extern "C" void kernel_launch(void* const* d_in, const int* in_sizes, int n_in,
                              void* d_out, int out_size, void* d_ws, size_t ws_size,
                              hipStream_t stream) {
    const float* x    = (const float*)d_in[0];
    const int*   edge = (const int*)d_in[1];
    const float* pos  = (const float*)d_in[2];
    const float* Wg   = (const float*)d_in[3];
    const float* bg   = (const float*)d_in[4];
    const float* Wp   = (const float*)d_in[5];
    const float* bp   = (const float*)d_in[6];

    const int N = in_sizes[0] / D;
    const int E = in_sizes[1] / 2;
    const int* srcIdx = edge;
    const int* dstIdx = edge + E;

    float* hs   = (float*)d_ws;                 // [N,64] scaled features (gather src)
    float* accb = hs   + (size_t)N * D;         // [N,64] aggregation (init = self loop)
    float* dinv = accb + (size_t)N * D;         // [N]    deg -> 1/sqrt(deg)

    float* out = (float*)d_out;

    const int nTiles = N / ROWS_PER_BLK;
    const int rem    = N - nTiles * ROWS_PER_BLK;
    const int remRow = nTiles * ROWS_PER_BLK;

    // 1) degree with self loops, then dinv = rsqrt(deg)
    k_init_deg<<<(N + 255) / 256, 256, 0, stream>>>(dinv, N);
    k_count_deg<<<(E + 255) / 256, 256, 0, stream>>>(dstIdx, dinv, E);
    k_dinv<<<(N + 255) / 256, 256, 0, stream>>>(dinv, N);

    // 2) hs = (x @ W_gcn) * dinv ; acc = hs (self loop)
    if (nTiles > 0)
        k_gemm_wmma<0><<<nTiles, BLK, 0, stream>>>(x, Wg, dinv, hs, accb,
                                                   nullptr, nullptr, nullptr);
    if (rem > 0)
        k_gemm_naive<0><<<(rem * D + 255) / 256, 256, 0, stream>>>(
            x, Wg, dinv, hs, accb, nullptr, nullptr, nullptr, remRow, rem);

    // 3) acc[dst] += hs[src] over all edges (one f32 atomic per element)
    {
        int total = E * D;
        k_scatter<<<(total + 255) / 256, 256, 0, stream>>>(srcIdx, dstIdx, hs, accb, total);
    }

    // 4) out = acc*dinv + (pos @ W_pos) + b_gcn + b_pos
    if (nTiles > 0)
        k_gemm_wmma<1><<<nTiles, BLK, 0, stream>>>(pos, Wp, dinv, out, nullptr,
                                                   accb, bg, bp);
    if (rem > 0)
        k_gemm_naive<1><<<(rem * D + 255) / 256, 256, 0, stream>>>(
            pos, Wp, dinv, out, nullptr, accb, bg, bp, remRow, rem);
}